// PointNet_SA_module_basic_33071248179389
// MI455X (gfx1250) — compile-verified
//
#include <hip/hip_runtime.h>

typedef __attribute__((ext_vector_type(2))) float v2f;
typedef __attribute__((ext_vector_type(8))) float v8f;

#define BB   8
#define NN   8192
#define NP   1024
#define NS   32
#define DPT  64
#define CHN  (3 + DPT)   // 67

// ---------------------------------------------------------------------------
// Kernel 1: farthest point sampling. One block per batch, 256 threads.
// Each thread owns 32 points entirely in registers (coords + running min-dist),
// so each of the 1024 serial iterations is VALU-only + wave32 shuffle argmax.
// ---------------------------------------------------------------------------
__global__ __launch_bounds__(256)
void fps_kernel(const float* __restrict__ xyz, float* __restrict__ out_newxyz) {
    const int b    = blockIdx.x;
    const int tid  = threadIdx.x;
    const int lane = tid & 31;
    const int wid  = tid >> 5;
    const float* x = xyz + (size_t)b * NN * 3;

    float X[32], Y[32], Z[32], Dm[32];
#pragma unroll
    for (int k = 0; k < 32; ++k) {
        int n = k * 256 + tid;
        X[k]  = x[n * 3 + 0];
        Y[k]  = x[n * 3 + 1];
        Z[k]  = x[n * 3 + 2];
        Dm[k] = 1.0e10f;
    }

    __shared__ float wval[8];
    __shared__ int   wind[8];
    __shared__ int   sFar;

    int farthest = 0;
    for (int i = 0; i < NP; ++i) {
        const float cx = x[farthest * 3 + 0];
        const float cy = x[farthest * 3 + 1];
        const float cz = x[farthest * 3 + 2];
        if (tid == 0) {
            float* o = out_newxyz + ((size_t)b * NP + i) * 3;
            o[0] = cx; o[1] = cy; o[2] = cz;
        }
        float best = -1.0f;
        int   bidx = 0;
#pragma unroll
        for (int k = 0; k < 32; ++k) {
            float dx = X[k] - cx, dy = Y[k] - cy, dz = Z[k] - cz;
            float d  = dx * dx + dy * dy + dz * dz;
            float nd = fminf(Dm[k], d);
            Dm[k] = nd;
            if (nd > best) { best = nd; bidx = k * 256 + tid; }
        }
#pragma unroll
        for (int off = 16; off > 0; off >>= 1) {
            float ov = __shfl_down(best, off, 32);
            int   oi = __shfl_down(bidx, off, 32);
            if (ov > best || (ov == best && oi < bidx)) { best = ov; bidx = oi; }
        }
        if (lane == 0) { wval[wid] = best; wind[wid] = bidx; }
        __syncthreads();
        if (tid == 0) {
            float bb = wval[0]; int bi = wind[0];
#pragma unroll
            for (int w = 1; w < 8; ++w)
                if (wval[w] > bb || (wval[w] == bb && wind[w] < bi)) { bb = wval[w]; bi = wind[w]; }
            sFar = bi;
        }
        __syncthreads();
        farthest = sFar;
    }
}

// ---------------------------------------------------------------------------
// Kernel 2: ball query + grouping.
//  - distance tiles via V_WMMA_F32_16X16X4_F32, A = (-2qx,-2qy,-2qz,0) hoisted
//  - point chunks staged with double-buffered GLOBAL_LOAD_ASYNC_TO_LDS (ASYNCcnt)
//  - first-32-in-order index selection via wave32 ballot compaction
//  - fused gather/concat of new_points
// Grid: (NP/16, B). Block: 256 threads = 8 waves; block owns 16 query rows.
// ---------------------------------------------------------------------------
__global__ __launch_bounds__(256)
void group_kernel(const float* __restrict__ xyz, const float* __restrict__ pts,
                  const float* __restrict__ radius_p,
                  const float* __restrict__ newxyz,
                  float* __restrict__ out_newpoints) {
    const int tile = blockIdx.x;
    const int b    = blockIdx.y;
    const int tid  = threadIdx.x;
    const float r  = *radius_p;
    const float r2 = r * r;

    __shared__ float qa[16][4];        // (-2x,-2y,-2z,0) per query (WMMA A rows)
    __shared__ float qx[16], qy[16], qz[16], qn[16];
    __shared__ float pc[2][256][4];    // double-buffered staged point coords
    __shared__ float sdist[16][256];
    __shared__ int   slist[16][NS];
    __shared__ int   scount[16];

    const float* xb = xyz + (size_t)b * NN * 3;

    if (tid < 16) {
        const float* q = newxyz + ((size_t)b * NP + tile * 16 + tid) * 3;
        float a = q[0], bb = q[1], cc = q[2];
        qx[tid] = a; qy[tid] = bb; qz[tid] = cc;
        qn[tid] = a * a + bb * bb + cc * cc;
        qa[tid][0] = -2.0f * a; qa[tid][1] = -2.0f * bb;
        qa[tid][2] = -2.0f * cc; qa[tid][3] = 0.0f;
        scount[tid] = 0;
    }
    __syncthreads();

    const int lane = tid & 31;
    const int wid  = tid >> 5;
    const int hi   = lane >> 4;
    const int l16  = lane & 15;
    const int half = tid >> 4;
    const int k0   = hi * 2;

    // hoisted WMMA A-operand and per-lane row norms (constant across chunks)
    v2f aop = *(const v2f*)&qa[l16][k0];
    float qnr[8];
#pragma unroll
    for (int rr = 0; rr < 8; ++rr) qnr[rr] = qn[rr + hi * 8];

    const unsigned long long xbase = (unsigned long long)xb;

    // prologue: async-stage chunk 0 into buffer 0 (12B/point, direct to LDS)
    {
        unsigned ldsa = (unsigned)(unsigned long long)&pc[0][tid][0];
        unsigned goff = (unsigned)(tid * 12);
        asm volatile("global_load_async_to_lds_b64 %0, %1, %2"
                     :: "v"(ldsa), "v"(goff), "s"(xbase) : "memory");
        asm volatile("global_load_async_to_lds_b32 %0, %1, %2 offset:8"
                     :: "v"(ldsa), "v"(goff), "s"(xbase) : "memory");
    }

    int  cnt  = 0;
    bool done = false;
    const int NCH = NN / 256;

    for (int c = 0; c < NCH && !done; ++c) {
        const int buf = c & 1;
        if (c + 1 < NCH) {
            // prefetch next chunk into the other buffer, then wait for current
            unsigned ldsa = (unsigned)(unsigned long long)&pc[buf ^ 1][tid][0];
            unsigned goff = (unsigned)(((c + 1) * 256 + tid) * 12);
            asm volatile("global_load_async_to_lds_b64 %0, %1, %2"
                         :: "v"(ldsa), "v"(goff), "s"(xbase) : "memory");
            asm volatile("global_load_async_to_lds_b32 %0, %1, %2 offset:8"
                         :: "v"(ldsa), "v"(goff), "s"(xbase) : "memory");
            asm volatile("s_wait_asynccnt 0x2" ::: "memory");
        } else {
            asm volatile("s_wait_asynccnt 0x0" ::: "memory");
        }
        __syncthreads();   // current buffer visible to all waves

        const int jb = c * 256;

        // each wave computes two 16x16 distance tiles with f32 WMMA (K=4)
#pragma unroll
        for (int t2 = 0; t2 < 2; ++t2) {
            const int t   = wid * 2 + t2;
            const int col = t * 16 + l16;
            float4 pcv = *(const float4*)&pc[buf][col][0];   // one ds_load_b128
            v2f bv;
            bv.x = hi ? pcv.z : pcv.x;
            bv.y = hi ? 0.0f  : pcv.y;
            float pn = pcv.x * pcv.x + pcv.y * pcv.y + pcv.z * pcv.z;
            v8f acc = {0.f,0.f,0.f,0.f,0.f,0.f,0.f,0.f};
            acc = __builtin_amdgcn_wmma_f32_16x16x4_f32(
                      false, aop, false, bv, (short)0, acc, false, false);
#pragma unroll
            for (int rr = 0; rr < 8; ++rr)
                sdist[rr + hi * 8][col] = qnr[rr] + pn + acc[rr];
        }
        __syncthreads();

        // ordered compaction: half-wave `half` scans its query row in j order
#pragma unroll 1
        for (int s = 0; s < 16; ++s) {
            int  jl  = s * 16 + l16;
            bool inr = (sdist[half][jl] <= r2);
            unsigned mask = (unsigned)__ballot(inr);
            unsigned m16  = (mask >> (hi * 16)) & 0xFFFFu;
            int pre = __popc(m16 & ((1u << l16) - 1u));
            int pos = cnt + pre;
            if (inr && pos < NS) slist[half][pos] = jb + jl;
            cnt += __popc(m16);
            if (cnt > NS) cnt = NS;
        }
        if (l16 == 0) scount[half] = cnt;
        __syncthreads();

        done = true;
#pragma unroll
        for (int q = 0; q < 16; ++q) done = done && (scount[q] >= NS);
    }
    __syncthreads();

    // pad short lists with the first hit (reference: idx==N -> first)
    if (tid < 16) {
        int cq    = scount[tid];
        int first = (cq > 0) ? slist[tid][0] : (NN - 1);
        for (int s = cq; s < NS; ++s) slist[tid][s] = first;
    }
    __syncthreads();

    // fused gather + concat: new_points[b, qg, s, 0:3]=xyz[idx]-q, [3:67]=pts[idx]
    for (int e = tid; e < 16 * NS * CHN; e += 256) {
        int q   = e / (NS * CHN);
        int rem = e % (NS * CHN);
        int s   = rem / CHN;
        int ch  = rem % CHN;
        int idx = slist[q][s];
        float val;
        if (ch < 3) {
            float pv = xb[(size_t)idx * 3 + ch];
            float qv = (ch == 0) ? qx[q] : ((ch == 1) ? qy[q] : qz[q]);
            val = pv - qv;
        } else {
            val = pts[((size_t)b * NN + idx) * DPT + (ch - 3)];
        }
        out_newpoints[(((size_t)b * NP + tile * 16 + q) * NS + s) * CHN + ch] = val;
    }
}

extern "C" void kernel_launch(void* const* d_in, const int* in_sizes, int n_in,
                              void* d_out, int out_size, void* d_ws, size_t ws_size,
                              hipStream_t stream) {
    (void)in_sizes; (void)n_in; (void)out_size; (void)d_ws; (void)ws_size;
    const float* xyz    = (const float*)d_in[0];
    const float* pts    = (const float*)d_in[1];
    const float* radius = (const float*)d_in[3];   // python float scalar -> 1-elt array

    float* out           = (float*)d_out;
    float* out_newxyz    = out;                              // (8,1024,3)
    float* out_newpoints = out + (size_t)BB * NP * 3;        // (8,1024,32,67)

    fps_kernel<<<dim3(BB), dim3(256), 0, stream>>>(xyz, out_newxyz);
    group_kernel<<<dim3(NP / 16, BB), dim3(256), 0, stream>>>(
        xyz, pts, radius, out_newxyz, out_newpoints);
}